// DetectionLoss_30743375904796
// MI455X (gfx1250) — compile-verified
//
#include <hip/hip_runtime.h>

#define GRID13     13
#define NUM_CLASS  20
#define NUM_ANCHOR 5
#define CHN        25          // 5 + NUM_CLASS
#define BATCH      512
#define MGT        32
#define PLANE      169         // 13*13
#define PER_BATCH  21125       // 125*169
#define TOTAL_EL   10816000    // 512*21125
#define BPG        32          // batches per thread in k_sumsq
#define NBG        (BATCH / BPG)   // 16 batch groups

typedef __attribute__((ext_vector_type(2))) float v2f;
typedef __attribute__((ext_vector_type(8))) float v8f;

// ANCHORS[0::2]/13 and ANCHORS[1::2]/13 (4 pairs; anchor 4 clamps to pair 3).
// Pure compare/select chains -> v_cndmask immediates, no memory traffic.
__device__ __forceinline__ float anchor_sw(int a) {
  return (a == 0) ? (1.3221f  / 13.0f)
       : (a == 1) ? (3.19275f / 13.0f)
       : (a == 2) ? (5.05587f / 13.0f)
       :            (9.47112f / 13.0f);
}
__device__ __forceinline__ float anchor_sh(int a) {
  return (a == 0) ? (1.73145f / 13.0f)
       : (a == 1) ? (4.00944f / 13.0f)
       : (a == 2) ? (8.09892f / 13.0f)
       :            (4.84053f / 13.0f);
}

__global__ void k_zero(float* acc) { acc[0] = 0.0f; }

// ---------------------------------------------------------------------------
// Kernel 1: Σ_all (scale_c * raw)^2.  Transposed mapping: thread owns one
// within-batch position j (fixed anchor/channel => scale^2 computed ONCE,
// as immediates), loops over 32 batches with stride 21125 floats.  All 32
// byte offsets (<= 2.70 MB) fit the signed-24-bit IOFFSET, so the unrolled
// body is 32x { global_load_b32 base, off imm } + 32 FMAs: pure HBM stream,
// ~43.3 MB -> ~1.9 us floor at 23.3 TB/s.
// Wave reduction done exactly with one V_WMMA_F32_16X16X4_F32:
//   A[m][0]=s_m (lanes 0-15), A[m][2]=s_{m+16} (lanes 16-31), B=ones, C=0
//   => D[m][n] = s_m + s_{m+16}; lane0 sums rows 0-7, lane16 rows 8-15.
// ---------------------------------------------------------------------------
__global__ void k_sumsq(const float* __restrict__ det, float* __restrict__ acc) {
  const int j = blockIdx.x * blockDim.x + threadIdx.x;   // position within batch
  float s = 0.0f;
  if (j < PER_BATCH) {
    const int ac = j / PLANE;          // a*25 + c, computed once per thread
    const int a  = ac / CHN;
    const int c  = ac - a * CHN;
    const float sw = anchor_sw(a);
    const float sh = anchor_sh(a);
    const float s2 = (c == 2) ? sw * sw : ((c == 3) ? sh * sh : 1.0f);

    const float* __restrict__ p = det + (size_t)blockIdx.y * (BPG * PER_BATCH) + j;
#pragma unroll
    for (int b = 0; b < BPG; ++b) {
      const float x = p[b * PER_BATCH];
      s = fmaf(s2 * x, x, s);
    }
  }

  // ---- exact wave32 reduction via WMMA (all lanes reconverged: EXEC ones) ----
  v2f a2; a2[0] = s;    a2[1] = 0.0f;
  v2f b2; b2[0] = 1.0f; b2[1] = 1.0f;   // all-ones B is layout-invariant
  v8f c2 = {};
  v8f d2 = __builtin_amdgcn_wmma_f32_16x16x4_f32(
      /*neg_a=*/false, a2, /*neg_b=*/false, b2,
      /*c_mod=*/(short)0, c2, /*reuse_a=*/false, /*reuse_b=*/false);
  float t = d2[0] + d2[1] + d2[2] + d2[3] + d2[4] + d2[5] + d2[6] + d2[7];

  __shared__ float red[16];            // 8 waves * 2 half-sums
  const int lane = threadIdx.x & 31;
  const int wave = threadIdx.x >> 5;
  if (lane == 0)  red[wave * 2 + 0] = t;   // rows 0-7  = s_0..7  + s_16..23
  if (lane == 16) red[wave * 2 + 1] = t;   // rows 8-15 = s_8..15 + s_24..31
  __syncthreads();
  if (threadIdx.x == 0) {
    float tot = 0.0f;
#pragma unroll
    for (int i = 0; i < 16; ++i) tot += red[i];
    atomicAdd(acc, tot);
  }
}

// ---------------------------------------------------------------------------
// Kernel 2: one wave per batch, one lane per GT box (M=32 == wave32).
// Gather pred (5 anchors x 4 ch), IoU, argmax, LDS duplicate resolution
// (highest m with same (best,ix,iy) wins == last-writer scatter set), then
// sparse correction Σ_c ((w_c g_c - d_c)^2 - d_c^2): only 6 nonzero channels.
// ---------------------------------------------------------------------------
__global__ void k_assign(const float* __restrict__ det,
                         const float* __restrict__ gt_xywh,
                         const int*   __restrict__ gt_class,
                         float* __restrict__ acc) {
  const int b = blockIdx.x;
  const int m = threadIdx.x;           // 0..31

  const float x = gt_xywh[((b * MGT) + m) * 4 + 0];
  const float y = gt_xywh[((b * MGT) + m) * 4 + 1];
  const float w = gt_xywh[((b * MGT) + m) * 4 + 2];
  const float h = gt_xywh[((b * MGT) + m) * 4 + 3];
  const int  cls = gt_class[b * MGT + m];

  int ix = (int)(x * (float)GRID13); ix = ix < 0 ? 0 : (ix > GRID13 - 1 ? GRID13 - 1 : ix);
  int iy = (int)(y * (float)GRID13); iy = iy < 0 ? 0 : (iy > GRID13 - 1 ? GRID13 - 1 : iy);
  const int cell   = ix * GRID13 + iy;
  const int base_b = b * PER_BATCH;

  int   best     = 0;
  float best_iou = -3.0e38f;
#pragma unroll
  for (int a = 0; a < NUM_ANCHOR; ++a) {
    const int pb = base_b + a * (CHN * PLANE) + cell;
    const float px = det[pb + 0 * PLANE];
    const float py = det[pb + 1 * PLANE];
    const float pw = det[pb + 2 * PLANE] * anchor_sw(a);
    const float ph = det[pb + 3 * PLANE] * anchor_sh(a);
    const float ix1 = fmaxf(x, px), iy1 = fmaxf(y, py);
    const float ix2 = fminf(w, pw), iy2 = fminf(h, ph);
    const float inter = (ix2 - ix1 + 1.0f) * (iy2 - iy1 + 1.0f);
    const float a1 = (w - x + 1.0f) * (h - y + 1.0f);
    const float a2 = (pw - px + 1.0f) * (ph - py + 1.0f);
    const float iou = inter / (a1 + a2 - inter);
    if (iou > best_iou) { best_iou = iou; best = a; }   // first max wins (strict >)
  }

  // duplicate resolution: highest m with identical target cell survives
  __shared__ int keys[MGT];
  const int key = best * PLANE + cell;
  keys[m] = key;
  __syncthreads();
  bool winner = true;
  for (int j = m + 1; j < MGT; ++j)
    if (keys[j] == key) { winner = false; break; }

  if (winner) {
    const int pb = base_b + best * (CHN * PLANE) + cell;
    const float d0 = det[pb + 0 * PLANE];
    const float d1 = det[pb + 1 * PLANE];
    const float d2v = det[pb + 2 * PLANE] * anchor_sw(best);
    const float d3v = det[pb + 3 * PLANE] * anchor_sh(best);
    const float d4 = det[pb + 4 * PLANE];
    const float dc = det[pb + (5 + cls) * PLANE];
    float corr = 0.0f;
    // xywh: weight 5, target g  -> (5g - d)^2 - d^2
    corr += (5.0f * x - d0)  * (5.0f * x - d0)  - d0  * d0;
    corr += (5.0f * y - d1)  * (5.0f * y - d1)  - d1  * d1;
    corr += (5.0f * w - d2v) * (5.0f * w - d2v) - d2v * d2v;
    corr += (5.0f * h - d3v) * (5.0f * h - d3v) - d3v * d3v;
    // objectness: weight 1 (obj_w==1 branch), target 1
    corr += (1.0f - d4) * (1.0f - d4) - d4 * d4;
    // one-hot class: weight 1, target 1 (other 19 classes contribute 0)
    corr += (1.0f - dc) * (1.0f - dc) - dc * dc;
    atomicAdd(acc, corr);
  }
}

__global__ void k_final(const float* __restrict__ acc, float* __restrict__ out) {
  out[0] = acc[0] * (1.0f / (float)TOTAL_EL);
}

extern "C" void kernel_launch(void* const* d_in, const int* in_sizes, int n_in,
                              void* d_out, int out_size, void* d_ws, size_t ws_size,
                              hipStream_t stream) {
  const float* det     = (const float*)d_in[0];   // (512,125,13,13) f32
  const float* gt_xywh = (const float*)d_in[1];   // (512,32,4)      f32
  const int*   gt_cls  = (const int*)  d_in[2];   // (512,32)        i32
  float* acc = (float*)d_ws;                      // scalar accumulator
  float* out = (float*)d_out;

  k_zero<<<1, 1, 0, stream>>>(acc);
  // x: ceil(21125/256) = 83 blocks over within-batch positions
  // y: 16 batch groups of 32 batches each  -> 340K threads, ~10.6K waves
  dim3 grid((PER_BATCH + 255) / 256, NBG);
  k_sumsq<<<grid, 256, 0, stream>>>(det, acc);
  k_assign<<<BATCH, MGT, 0, stream>>>(det, gt_xywh, gt_cls, acc);
  k_final<<<1, 1, 0, stream>>>(acc, out);
}